// Dual_axis_21792664059969
// MI455X (gfx1250) — compile-verified
//
#include <hip/hip_runtime.h>

// ---- problem constants ----
#define Bb   32
#define Cc   128
#define Nn   207
#define Tt   12
#define Hh   4
#define HID  32
#define HD   8
#define NT   (Nn * Tt)        // 2484
#define GG   (Hh * Bb)        // 128
#define NTOK (Bb * NT)        // 79488 = 16 * 4968
#define MTILES (NTOK / 16)    // 4968

typedef float v2f __attribute__((ext_vector_type(2)));
typedef float v8f __attribute__((ext_vector_type(8)));

#define SCALE 0.35355339059327373f   // 1/sqrt(8)

// ============================================================================
// Kernel 1: fused projections nq/nk/tq/tk/v1 = x @ W^T + b  (WMMA f32 16x16x4)
// grid (4968 M-tiles, 10 output-tiles of 16), one wave per block.
// x: [B, C, N, T].  Outputs laid out for their consumers:
//   nq,nk: [B, T, H, N, HD]   tq,tk: [B, N, H, T, HD]   v1: [H*B, HD, N, T]
// ============================================================================
__global__ __launch_bounds__(32) void proj_kernel(
    const float* __restrict__ x,
    const float* __restrict__ Wnq, const float* __restrict__ bnq, float* __restrict__ nq,
    const float* __restrict__ Wnk, const float* __restrict__ bnk, float* __restrict__ nk,
    const float* __restrict__ Wtq, const float* __restrict__ btq, float* __restrict__ tq,
    const float* __restrict__ Wtk, const float* __restrict__ btk, float* __restrict__ tk,
    const float* __restrict__ Wv,  const float* __restrict__ bv,  float* __restrict__ v1)
{
    const int lane = threadIdx.x;
    const int mt   = blockIdx.x;        // token tile
    const int ot   = blockIdx.y;        // 0..9: which 16 of the 160 fused outputs
    const int mat  = ot >> 1;           // 0 nq, 1 nk, 2 tq, 3 tk, 4 v
    const int ob   = (ot & 1) * 16;     // output-row base within the matrix

    const float* W;  const float* bias;
    switch (mat) {
        case 0:  W = Wnq; bias = bnq; break;
        case 1:  W = Wnk; bias = bnk; break;
        case 2:  W = Wtq; bias = btq; break;
        case 3:  W = Wtk; bias = btk; break;
        default: W = Wv;  bias = bv;  break;
    }

    // ---- A fragment addressing: lane&15 = token row, lane-half picks K pair ----
    const int nl = lane & 15;
    const int kb = (lane >> 4) * 2;
    const int m  = mt * 16 + nl;
    const int ab = m / NT;
    const int an = m - ab * NT;
    const float* xrow = x + (size_t)ab * Cc * NT + an;     // + k*NT per channel
    const float* Wr   = W + (size_t)(ob + nl) * Cc;        // B[k][n] = W[ob+n][k]

    v8f acc = {0.f,0.f,0.f,0.f,0.f,0.f,0.f,0.f};
    #pragma unroll
    for (int k0 = 0; k0 < Cc; k0 += 4) {
        v2f a, b;
        a.x = xrow[(size_t)(k0 + kb)     * NT];
        a.y = xrow[(size_t)(k0 + kb + 1) * NT];
        b.x = Wr[k0 + kb];
        b.y = Wr[k0 + kb + 1];
        acc = __builtin_amdgcn_wmma_f32_16x16x4_f32(false, a, false, b,
                                                    (short)0, acc, false, false);
    }

    // ---- D layout: VGPR i holds rows {i, i+8}; column = lane&15 ----
    const int hiHalf = (lane >> 4) * 8;
    const int o = ob + nl;              // 0..31 within matrix
    const float bi = bias[o];
    const int h = o >> 3, d = o & 7;
    #pragma unroll
    for (int i = 0; i < 8; i++) {
        const int ms  = mt * 16 + hiHalf + i;
        const int b2  = ms / NT;
        const int nt2 = ms - b2 * NT;
        const int n   = nt2 / Tt;
        const int t   = nt2 - n * Tt;
        const float val = acc[i] + bi;
        if (mat <= 1) {
            size_t idx = ((((size_t)b2 * Tt + t) * Hh + h) * Nn + n) * HD + d;
            (mat == 0 ? nq : nk)[idx] = val;
        } else if (mat <= 3) {
            size_t idx = ((((size_t)b2 * Nn + n) * Hh + h) * Tt + t) * HD + d;
            (mat == 2 ? tq : tk)[idx] = val;
        } else {
            size_t idx = (((size_t)(h * Bb + b2) * HD + d) * Nn + n) * Tt + t;
            v1[idx] = val;
        }
    }
}

// ============================================================================
// Kernel 2: temporal attention (12x12 softmax) fused with temporal graph conv.
// One thread per (g, n), g = flat fused index (replicates reference's mixed
// head/batch-major pairing: attn decoded as b=g/H, h=g%H; v1 read at flat g).
// ht[g, c, n, j] = sum_i v1[g, c, n, i] * P[i][j]
// ============================================================================
__global__ __launch_bounds__(256) void temporal_kernel(
    const float* __restrict__ tq, const float* __restrict__ tk,
    const float* __restrict__ v1, const float* __restrict__ Bt,
    float* __restrict__ ht)
{
    const int idx = blockIdx.x * 256 + threadIdx.x;
    if (idx >= GG * Nn) return;
    const int g = idx / Nn, n = idx - g * Nn;
    const int b2 = g >> 2, h2 = g & 3;

    const float* tqb = tq + (((size_t)b2 * Nn + n) * Hh + h2) * Tt * HD;
    const float* tkb = tk + (((size_t)b2 * Nn + n) * Hh + h2) * Tt * HD;
    const float* Btb = Bt + (size_t)h2 * Tt * Tt;

    float P[Tt][Tt];
    #pragma unroll
    for (int i = 0; i < Tt; i++) {
        float qi[HD];
        #pragma unroll
        for (int d = 0; d < HD; d++) qi[d] = tqb[i * HD + d];
        float mx = -1e30f;
        #pragma unroll
        for (int j = 0; j < Tt; j++) {
            float s = 0.f;
            #pragma unroll
            for (int d = 0; d < HD; d++) s += qi[d] * tkb[j * HD + d];
            s = s * SCALE + Btb[i * Tt + j];
            P[i][j] = s;
            mx = fmaxf(mx, s);
        }
        float sum = 0.f;
        #pragma unroll
        for (int j = 0; j < Tt; j++) { float e = __expf(P[i][j] - mx); P[i][j] = e; sum += e; }
        const float inv = 1.f / sum;
        #pragma unroll
        for (int j = 0; j < Tt; j++) P[i][j] *= inv;
    }
    #pragma unroll
    for (int c = 0; c < HD; c++) {
        const float* vr = v1 + (((size_t)g * HD + c) * Nn + n) * Tt;
        float vv[Tt];
        #pragma unroll
        for (int i = 0; i < Tt; i++) vv[i] = vr[i];
        float* op = ht + (((size_t)g * HD + c) * Nn + n) * Tt;
        #pragma unroll
        for (int j = 0; j < Tt; j++) {
            float s = 0.f;
            #pragma unroll
            for (int i = 0; i < Tt; i++) s += vv[i] * P[i][j];
            op[j] = s;
        }
    }
}

// ============================================================================
// Kernel 3: spatial attention fused with spatial graph conv; never
// materializes the 263MB attention tensor. One block per (g, t):
//  - 16 query rows per chunk: logits + wave-wide softmax (wave32 shfl_xor),
//    * adjacency mask -> 13KB LDS tile
//  - rank-16 update of the 8x207 conv accumulator (thread tid owns column tid)
// ============================================================================
__global__ __launch_bounds__(256) void spatial_kernel(
    const float* __restrict__ nq, const float* __restrict__ nk,
    const float* __restrict__ v1, const float* __restrict__ adj,
    const float* __restrict__ Bn, float* __restrict__ hs)
{
    __shared__ float nq_s[Nn * HD];
    __shared__ float nk_s[Nn * HD];
    __shared__ float v1_s[HD * Nn];
    __shared__ float A_s[16 * Nn];

    const int g = blockIdx.x, l = blockIdx.y;
    const int b2 = g >> 2, h2 = g & 3;
    const int tid = threadIdx.x;
    const int wv = tid >> 5, lane = tid & 31;

    const float* nqg = nq + (((size_t)b2 * Tt + l) * Hh + h2) * (size_t)Nn * HD;
    const float* nkg = nk + (((size_t)b2 * Tt + l) * Hh + h2) * (size_t)Nn * HD;
    for (int e = tid; e < Nn * HD; e += 256) { nq_s[e] = nqg[e]; nk_s[e] = nkg[e]; }
    for (int e = tid; e < HD * Nn; e += 256) {
        int c = e / Nn, v = e - c * Nn;
        v1_s[e] = v1[(((size_t)g * HD + c) * Nn + v) * Tt + l];
    }
    __syncthreads();

    const float* adjb = adj + ((size_t)b2 * Tt + l) * (size_t)Nn * Nn;
    const float* Bnb  = Bn + (size_t)h2 * Nn * Nn;

    float acc[HD];
    #pragma unroll
    for (int c = 0; c < HD; c++) acc[c] = 0.f;

    for (int v0 = 0; v0 < Nn; v0 += 16) {
        // ---- phase A: 16 logit rows, wave per 2 rows, softmax over w ----
        #pragma unroll
        for (int r = 0; r < 2; r++) {
            const int v = v0 + wv * 2 + r;
            if (v < Nn) {
                float lg[7];
                float mx = -1e30f;
                #pragma unroll
                for (int j = 0; j < 7; j++) {
                    const int w = lane + 32 * j;
                    float s = -1e30f;
                    if (w < Nn) {
                        float t = 0.f;
                        #pragma unroll
                        for (int d = 0; d < HD; d++)
                            t += nq_s[v * HD + d] * nk_s[w * HD + d];
                        s = t * SCALE + Bnb[(size_t)v * Nn + w];
                    }
                    lg[j] = s;
                    mx = fmaxf(mx, s);
                }
                #pragma unroll
                for (int off = 16; off > 0; off >>= 1)
                    mx = fmaxf(mx, __shfl_xor(mx, off, 32));
                float sum = 0.f;
                #pragma unroll
                for (int j = 0; j < 7; j++) {
                    const int w = lane + 32 * j;
                    const float e = (w < Nn) ? __expf(lg[j] - mx) : 0.f;
                    lg[j] = e; sum += e;
                }
                #pragma unroll
                for (int off = 16; off > 0; off >>= 1)
                    sum += __shfl_xor(sum, off, 32);
                const float inv = 1.f / sum;
                #pragma unroll
                for (int j = 0; j < 7; j++) {
                    const int w = lane + 32 * j;
                    if (w < Nn)
                        A_s[(v - v0) * Nn + w] = lg[j] * inv * adjb[(size_t)v * Nn + w];
                }
            }
        }
        __syncthreads();
        // ---- phase B: acc[c, w=tid] += A[vi, w] * v1[c, vi] ----
        if (tid < Nn) {
            const int vmax = (Nn - v0 < 16) ? (Nn - v0) : 16;
            for (int vi = 0; vi < vmax; vi++) {
                const float a = A_s[vi * Nn + tid];
                #pragma unroll
                for (int c = 0; c < HD; c++) acc[c] += a * v1_s[c * Nn + v0 + vi];
            }
        }
        __syncthreads();
    }
    if (tid < Nn) {
        #pragma unroll
        for (int c = 0; c < HD; c++)
            hs[(((size_t)g * HD + c) * Nn + tid) * Tt + l] = acc[c];
    }
}

// ============================================================================
// Kernel 4: v = W_fcv@x + b; out = W_proj@(h*v) + b  (both via WMMA f32).
// One block per 16-token tile: phase0 stage X tile to LDS, phase1 (4 waves)
// compute 16x64 hv = (fcv result)*h into LDS, phase2 (8 waves) 16x128 out.
// ============================================================================
__global__ __launch_bounds__(256) void out_kernel(
    const float* __restrict__ x,
    const float* __restrict__ Wfcv,  const float* __restrict__ bfcv,
    const float* __restrict__ Wproj, const float* __restrict__ bproj,
    const float* __restrict__ hs, const float* __restrict__ ht,
    float* __restrict__ out)
{
    __shared__ float X_s[16 * Cc];
    __shared__ float HV_s[16 * 64];

    const int tid = threadIdx.x;
    const int m0 = blockIdx.x * 16;

    for (int e = tid; e < 16 * Cc; e += 256) {
        const int ml = e & 15, k = e >> 4;
        const int m = m0 + ml;
        const int bb = m / NT, nt = m - bb * NT;
        X_s[ml * Cc + k] = x[((size_t)bb * Cc + k) * NT + nt];
    }
    __syncthreads();

    const int wv = tid >> 5, lane = tid & 31;
    const int nl = lane & 15;
    const int kb = (lane >> 4) * 2;
    const int hiHalf = (lane >> 4) * 8;

    if (wv < 4) {   // wave-uniform branch: EXEC stays all-1s for WMMA
        const int o0 = wv * 16;
        const float* Wr = Wfcv + (size_t)(o0 + nl) * Cc;
        v8f acc = {0.f,0.f,0.f,0.f,0.f,0.f,0.f,0.f};
        #pragma unroll
        for (int k0 = 0; k0 < Cc; k0 += 4) {
            v2f a, b;
            a.x = X_s[nl * Cc + k0 + kb];
            a.y = X_s[nl * Cc + k0 + kb + 1];
            b.x = Wr[k0 + kb];
            b.y = Wr[k0 + kb + 1];
            acc = __builtin_amdgcn_wmma_f32_16x16x4_f32(false, a, false, b,
                                                        (short)0, acc, false, false);
        }
        const int o = o0 + nl;
        const float bi = bfcv[o];
        const int hh = o >> 4, c2 = o & 15;     // h = cat([ht(8), hs(8)]) per head
        #pragma unroll
        for (int i = 0; i < 8; i++) {
            const int ml = hiHalf + i;
            const int m = m0 + ml;
            const int bb = m / NT, nt = m - bb * NT;
            const int n = nt / Tt, t = nt - n * Tt;
            const int g = hh * Bb + bb;         // head-major v1/hs/ht index
            float hval;
            if (c2 < 8) hval = ht[(((size_t)g * HD + c2)       * Nn + n) * Tt + t];
            else        hval = hs[(((size_t)g * HD + (c2 - 8)) * Nn + n) * Tt + t];
            HV_s[ml * 64 + o] = (acc[i] + bi) * hval;
        }
    }
    __syncthreads();

    {
        const int co0 = wv * 16;
        const float* Wr = Wproj + (size_t)(co0 + nl) * 64;
        v8f acc = {0.f,0.f,0.f,0.f,0.f,0.f,0.f,0.f};
        #pragma unroll
        for (int k0 = 0; k0 < 64; k0 += 4) {
            v2f a, b;
            a.x = HV_s[nl * 64 + k0 + kb];
            a.y = HV_s[nl * 64 + k0 + kb + 1];
            b.x = Wr[k0 + kb];
            b.y = Wr[k0 + kb + 1];
            acc = __builtin_amdgcn_wmma_f32_16x16x4_f32(false, a, false, b,
                                                        (short)0, acc, false, false);
        }
        const int co = co0 + nl;
        const float bi = bproj[co];
        #pragma unroll
        for (int i = 0; i < 8; i++) {
            const int ml = hiHalf + i;
            const int m = m0 + ml;
            const int bb = m / NT, nt = m - bb * NT;
            out[((size_t)bb * Cc + co) * NT + nt] = acc[i] + bi;
        }
    }
}

// ============================================================================
extern "C" void kernel_launch(void* const* d_in, const int* in_sizes, int n_in,
                              void* d_out, int out_size, void* d_ws, size_t ws_size,
                              hipStream_t stream)
{
    (void)in_sizes; (void)n_in; (void)out_size; (void)ws_size;
    const float* x     = (const float*)d_in[0];
    const float* adj   = (const float*)d_in[1];
    // d_in[2..4] (time_ind/weekin/holiin) unused by the reference math
    const float* Wnq = (const float*)d_in[5],  *bnq = (const float*)d_in[6];
    const float* Wnk = (const float*)d_in[7],  *bnk = (const float*)d_in[8];
    const float* Wtq = (const float*)d_in[9],  *btq = (const float*)d_in[10];
    const float* Wtk = (const float*)d_in[11], *btk = (const float*)d_in[12];
    const float* Wv  = (const float*)d_in[13], *bv  = (const float*)d_in[14];
    const float* Wfcv = (const float*)d_in[15], *bfcv = (const float*)d_in[16];
    const float* Wproj= (const float*)d_in[17], *bproj= (const float*)d_in[18];
    const float* Bn  = (const float*)d_in[19];
    const float* Bt  = (const float*)d_in[20];
    float* out = (float*)d_out;

    // workspace: 7 equal segments of 2,543,616 f32 (~71 MB total, L2-resident)
    const size_t SEG = (size_t)GG * HD * Nn * Tt;   // == B*T*H*N*HD
    float* ws  = (float*)d_ws;
    float* nq = ws;           float* nk = ws + SEG;
    float* tq = ws + 2*SEG;   float* tk = ws + 3*SEG;
    float* v1 = ws + 4*SEG;
    float* hs = ws + 5*SEG;   float* ht = ws + 6*SEG;

    proj_kernel<<<dim3(MTILES, 10), 32, 0, stream>>>(
        x, Wnq, bnq, nq, Wnk, bnk, nk, Wtq, btq, tq, Wtk, btk, tk, Wv, bv, v1);

    temporal_kernel<<<(GG * Nn + 255) / 256, 256, 0, stream>>>(tq, tk, v1, Bt, ht);

    spatial_kernel<<<dim3(GG, Tt), 256, 0, stream>>>(nq, nk, v1, adj, Bn, hs);

    out_kernel<<<MTILES, 256, 0, stream>>>(x, Wfcv, bfcv, Wproj, bproj, hs, ht, out);
}